// LNN_38998303047829
// MI455X (gfx1250) — compile-verified
//
#include <hip/hip_runtime.h>

typedef __attribute__((ext_vector_type(2))) float v2f;
typedef __attribute__((ext_vector_type(8))) float v8f;

#define DIM  2048
#define HID  4096
#define FREE 1024
#define NS_ITERS 32

#define TM 64
#define TN 64
#define TK 32

__device__ __forceinline__ float sigmoidf_(float x) { return 1.0f / (1.0f + expf(-x)); }
__device__ __forceinline__ float softplusf_(float x) {
    if (x > 20.0f) return x;
    return log1pf(expf(x));
}

__device__ __forceinline__ unsigned lds_off_u32(const void* p) {
    // __shared__ flat address: low 32 bits are the LDS byte offset (aperture mapping)
    return (unsigned)(size_t)p;
}

__device__ __forceinline__ void async_b64(unsigned lds_byte, const float* g) {
    asm volatile("global_load_async_to_lds_b64 %0, %1, off" :: "v"(lds_byte), "v"(g) : "memory");
}
__device__ __forceinline__ void async_b32(unsigned lds_byte, const float* g) {
    asm volatile("global_load_async_to_lds_b32 %0, %1, off" :: "v"(lds_byte), "v"(g) : "memory");
}

__device__ __forceinline__ float blockReduceSum256(float v) {
    __shared__ float sm[256];
    int t = threadIdx.x;
    sm[t] = v;
    __syncthreads();
    for (int s = 128; s > 0; s >>= 1) {
        if (t < s) sm[t] += sm[t + s];
        __syncthreads();
    }
    return sm[0];
}

// ---- layer 1: a1 = x@W1 + b1 ; h1 = softplus(a1) ; s1 = sigmoid(a1) ----
__global__ void k_layer1(const float* __restrict__ x, const float* __restrict__ W1,
                         const float* __restrict__ b1,
                         float* __restrict__ h1, float* __restrict__ s1) {
    int j = blockIdx.x * blockDim.x + threadIdx.x;
    if (j >= HID) return;
    float acc = b1[j];
    for (int i = 0; i < DIM; ++i) acc = fmaf(x[i], W1[(size_t)i * HID + j], acc);
    h1[j] = softplusf_(acc);
    s1[j] = sigmoidf_(acc);
}

// ---- layer 2: u = sigmoid(a2)*w3 ; cvec = sig*(1-sig)*w3 ----
__global__ void k_layer2(const float* __restrict__ h1, const float* __restrict__ W2,
                         const float* __restrict__ b2, const float* __restrict__ W3,
                         float* __restrict__ u, float* __restrict__ cvec) {
    int k = blockIdx.x * blockDim.x + threadIdx.x;
    if (k >= HID) return;
    float acc = b2[k];
    for (int j = 0; j < HID; ++j) acc = fmaf(h1[j], W2[(size_t)j * HID + k], acc);
    float s  = sigmoidf_(acc);
    float w3 = W3[k];
    u[k]    = s * w3;
    cvec[k] = s * (1.0f - s) * w3;
}

// ---- g2_j = W2[j,:]·u ; d1 = s1(1-s1)*g2 ; g1 = s1*g2 ----
__global__ void k_g2(const float* __restrict__ W2, const float* __restrict__ u,
                     const float* __restrict__ s1,
                     float* __restrict__ d1, float* __restrict__ g1) {
    int j = blockIdx.x;
    const float* row = W2 + (size_t)j * HID;
    float v = 0.f;
    for (int k = threadIdx.x; k < HID; k += blockDim.x) v = fmaf(row[k], u[k], v);
    float g2 = blockReduceSum256(v);
    if (threadIdx.x == 0) {
        float s = s1[j];
        d1[j] = s * (1.0f - s) * g2;
        g1[j] = s * g2;
    }
}

// ---- dxq_i = W1[i,:]·g1 (i < FREE) ----
__global__ void k_dx(const float* __restrict__ W1, const float* __restrict__ g1,
                     float* __restrict__ dxq) {
    int i = blockIdx.x;
    const float* row = W1 + (size_t)i * HID;
    float v = 0.f;
    for (int j = threadIdx.x; j < HID; j += blockDim.x) v = fmaf(row[j], g1[j], v);
    float r = blockReduceSum256(v);
    if (threadIdx.x == 0) dxq[i] = r;
}

// ---- out[r,c] = in[r,c] * scale[c] ----
__global__ void k_colscale(const float* __restrict__ in, const float* __restrict__ scale,
                           float* __restrict__ out, int cols, int total) {
    int i = blockIdx.x * blockDim.x + threadIdx.x;
    if (i < total) out[i] = in[i] * scale[i & (cols - 1)];
}

// LDS pair-layout: element (row, k) of a 64x32 tile lives at byte offset
//   (k>>1)*512 + row*8 + (k&1)*4
// -> each WMMA fragment (v2f of consecutive k) is one 8B DS load, conflict-free.

#define LOAD_FRAGS(buf, kk, A0, A1, B0, B1)                 \
    do {                                                    \
        const int p_  = ((kk) >> 1) + lh;                   \
        int oa0_ = p_ * 128 + (wm + l16) * 2;               \
        int oa1_ = p_ * 128 + (wm + 16 + l16) * 2;          \
        int ob0_ = p_ * 128 + (wn + l16) * 2;               \
        int ob1_ = p_ * 128 + (wn + 16 + l16) * 2;          \
        A0.x = sA[buf][oa0_]; A0.y = sA[buf][oa0_ + 1];     \
        A1.x = sA[buf][oa1_]; A1.y = sA[buf][oa1_ + 1];     \
        B0.x = sB[buf][ob0_]; B0.y = sB[buf][ob0_ + 1];     \
        B1.x = sB[buf][ob1_]; B1.y = sB[buf][ob1_ + 1];     \
    } while (0)

#define WMMA4(A0, A1, B0, B1)                                                                                    \
    do {                                                                                                         \
        acc[0][0] = __builtin_amdgcn_wmma_f32_16x16x4_f32(false, A0, false, B0, (short)0, acc[0][0], false, false); \
        acc[0][1] = __builtin_amdgcn_wmma_f32_16x16x4_f32(false, A0, false, B1, (short)0, acc[0][1], false, false); \
        acc[1][0] = __builtin_amdgcn_wmma_f32_16x16x4_f32(false, A1, false, B0, (short)0, acc[1][0], false, false); \
        acc[1][1] = __builtin_amdgcn_wmma_f32_16x16x4_f32(false, A1, false, B1, (short)0, acc[1][1], false, false); \
    } while (0)

// ======== async double-buffered WMMA f32 GEMM:  D = c0*Cin + c1*(A @ B^T) ========
// A row-major M×K, B row-major N×K. 16 b64 async copies per thread per chunk.
__global__ __launch_bounds__(128) void k_gemm_abt_async(
    const float* __restrict__ A, int lda,
    const float* __restrict__ B, int ldb,
    const float* Cin, int ldc, float c0, float c1,
    float* D, int ldd, int K) {
    __shared__ float sA[2][TM * TK];
    __shared__ float sB[2][TN * TK];
    const int m0   = blockIdx.y * TM;
    const int n0   = blockIdx.x * TN;
    const int tid  = threadIdx.x;
    const int lane = tid & 31;
    const int wave = tid >> 5;
    const int wm   = (wave >> 1) * 32;
    const int wn   = (wave & 1) * 32;
    const int lh   = lane >> 4;
    const int l16  = lane & 15;

    const unsigned ldsA = lds_off_u32(&sA[0][0]);
    const unsigned ldsB = lds_off_u32(&sB[0][0]);

    auto issue = [&](int buf, int k0) {
#pragma unroll
        for (int i = 0; i < 8; ++i) {
            int pidx = tid + i * 128;          // 1024 k-pairs per tile
            int p = pidx & 15;                 // k-pair (coalesced in global)
            int m = pidx >> 4;                 // row 0..63
            unsigned lo = (unsigned)(buf * 8192 + p * 512 + m * 8);
            async_b64(ldsA + lo, A + (size_t)(m0 + m) * lda + k0 + 2 * p);
            async_b64(ldsB + lo, B + (size_t)(n0 + m) * ldb + k0 + 2 * p);
        }
    };

    v8f acc[2][2] = {};
    const int nch = K / TK;
    issue(0, 0);
    for (int ch = 0; ch < nch; ++ch) {
        const int cur = ch & 1;
        const bool hasNext = (ch + 1 < nch);
        if (hasNext) {
            issue(cur ^ 1, (ch + 1) * TK);
            asm volatile("s_wait_asynccnt 0x10" ::: "memory"); // prev 16 done, next 16 in flight
        } else {
            asm volatile("s_wait_asynccnt 0x0" ::: "memory");
        }
        __syncthreads();
        // software-pipelined fragment consumption: load kk+4 before issuing kk's WMMAs
        v2f a0, a1, b0, b1;
        LOAD_FRAGS(cur, 0, a0, a1, b0, b1);
#pragma unroll
        for (int kk = 0; kk < TK; kk += 4) {
            v2f na0, na1, nb0, nb1;
            if (kk + 4 < TK) LOAD_FRAGS(cur, kk + 4, na0, na1, nb0, nb1);
            WMMA4(a0, a1, b0, b1);
            if (kk + 4 < TK) { a0 = na0; a1 = na1; b0 = nb0; b1 = nb1; }
        }
        __syncthreads();
    }
#pragma unroll
    for (int si = 0; si < 2; ++si)
#pragma unroll
        for (int sj = 0; sj < 2; ++sj)
#pragma unroll
            for (int r = 0; r < 8; ++r) {
                int row = m0 + wm + 16 * si + 8 * lh + r;
                int col = n0 + wn + 16 * sj + l16;
                float v = c1 * acc[si][sj][r];
                if (Cin) v = fmaf(c0, Cin[(size_t)row * ldc + col], v);
                D[(size_t)row * ldd + col] = v;
            }
}

// ======== async double-buffered WMMA f32 GEMM:  D = A @ B ========
// A row-major M×K (b64 copies), B row-major K×N (transposing b32 copies).
__global__ __launch_bounds__(128) void k_gemm_rr_async(
    const float* __restrict__ A, int lda,
    const float* __restrict__ B, int ldb,
    float* D, int ldd, int K) {
    __shared__ float sA[2][TM * TK];
    __shared__ float sB[2][TN * TK];
    const int m0   = blockIdx.y * TM;
    const int n0   = blockIdx.x * TN;
    const int tid  = threadIdx.x;
    const int lane = tid & 31;
    const int wave = tid >> 5;
    const int wm   = (wave >> 1) * 32;
    const int wn   = (wave & 1) * 32;
    const int lh   = lane >> 4;
    const int l16  = lane & 15;

    const unsigned ldsA = lds_off_u32(&sA[0][0]);
    const unsigned ldsB = lds_off_u32(&sB[0][0]);

    auto issue = [&](int buf, int k0) {
#pragma unroll
        for (int i = 0; i < 8; ++i) {
            int pidx = tid + i * 128;
            int p = pidx & 15, m = pidx >> 4;
            async_b64(ldsA + (unsigned)(buf * 8192 + p * 512 + m * 8),
                      A + (size_t)(m0 + m) * lda + k0 + 2 * p);
        }
#pragma unroll
        for (int i = 0; i < 16; ++i) {       // B: (k,n) -> pair layout, transposing scatter
            int idx = tid + i * 128;         // 2048 elements
            int n = idx & 63, k = idx >> 6;  // coalesced over n in global
            async_b32(ldsB + (unsigned)(buf * 8192 + (k >> 1) * 512 + n * 8 + (k & 1) * 4),
                      B + (size_t)(k0 + k) * ldb + n0 + n);
        }
    };

    v8f acc[2][2] = {};
    const int nch = K / TK;
    issue(0, 0);
    for (int ch = 0; ch < nch; ++ch) {
        const int cur = ch & 1;
        const bool hasNext = (ch + 1 < nch);
        if (hasNext) {
            issue(cur ^ 1, (ch + 1) * TK);
            asm volatile("s_wait_asynccnt 0x18" ::: "memory"); // 24 issued per chunk
        } else {
            asm volatile("s_wait_asynccnt 0x0" ::: "memory");
        }
        __syncthreads();
        v2f a0, a1, b0, b1;
        LOAD_FRAGS(cur, 0, a0, a1, b0, b1);
#pragma unroll
        for (int kk = 0; kk < TK; kk += 4) {
            v2f na0, na1, nb0, nb1;
            if (kk + 4 < TK) LOAD_FRAGS(cur, kk + 4, na0, na1, nb0, nb1);
            WMMA4(a0, a1, b0, b1);
            if (kk + 4 < TK) { a0 = na0; a1 = na1; b0 = nb0; b1 = nb1; }
        }
        __syncthreads();
    }
#pragma unroll
    for (int si = 0; si < 2; ++si)
#pragma unroll
        for (int sj = 0; sj < 2; ++sj)
#pragma unroll
            for (int r = 0; r < 8; ++r) {
                int row = m0 + wm + 16 * si + 8 * lh + r;
                int col = n0 + wn + 16 * sj + l16;
                D[(size_t)row * ldd + col] = acc[si][sj][r];
            }
}

// ---- rhs = dxq - Hv[:, :FREE] @ x[FREE:] ----
__global__ void k_rhs(const float* __restrict__ Hv, const float* __restrict__ x,
                      const float* __restrict__ dxq, float* __restrict__ rhs) {
    int r = blockIdx.x;
    const float* row = Hv + (size_t)r * DIM;
    float v = 0.f;
    for (int m = threadIdx.x; m < FREE; m += blockDim.x) v = fmaf(row[m], x[FREE + m], v);
    float s = blockReduceSum256(v);
    if (threadIdx.x == 0) rhs[r] = dxq[r] - s;
}

// ---- S = Hv[:, FREE:] ; rowsum = sum|S| ----
__global__ void k_extractS(const float* __restrict__ Hv, float* __restrict__ S,
                           float* __restrict__ rowsum) {
    int r = blockIdx.x;
    const float* row = Hv + (size_t)r * DIM + FREE;
    float v = 0.f;
    for (int m = threadIdx.x; m < FREE; m += blockDim.x) {
        float e = row[m];
        S[(size_t)r * FREE + m] = e;
        v += fabsf(e);
    }
    float s = blockReduceSum256(v);
    if (threadIdx.x == 0) rowsum[r] = s;
}

__global__ void k_alpha(const float* __restrict__ rowsum, float* __restrict__ alpha) {
    __shared__ float sm[256];
    int t = threadIdx.x;
    float v = 0.f;
    for (int i = t; i < FREE; i += 256) v = fmaxf(v, rowsum[i]);
    sm[t] = v;
    __syncthreads();
    for (int s = 128; s > 0; s >>= 1) {
        if (t < s) sm[t] = fmaxf(sm[t], sm[t + s]);
        __syncthreads();
    }
    if (t == 0) {
        float m = sm[0];
        alpha[0] = (m > 0.f) ? 1.0f / (m * m) : 0.f;
    }
}

__global__ void k_scaleX0(const float* __restrict__ S, const float* __restrict__ alpha,
                          float* __restrict__ X) {
    int i = blockIdx.x * blockDim.x + threadIdx.x;
    if (i < FREE * FREE) X[i] = S[i] * alpha[0];
}

__global__ void k_out(const float* __restrict__ X, const float* __restrict__ rhs,
                      float* __restrict__ out) {
    int r = blockIdx.x;
    float v = 0.f;
    for (int m = threadIdx.x; m < FREE; m += blockDim.x)
        v = fmaf(X[(size_t)r * FREE + m], rhs[m], v);
    float s = blockReduceSum256(v);
    if (threadIdx.x == 0) out[r] = s;
}

extern "C" void kernel_launch(void* const* d_in, const int* in_sizes, int n_in,
                              void* d_out, int out_size, void* d_ws, size_t ws_size,
                              hipStream_t stream) {
    (void)in_sizes; (void)n_in; (void)out_size; (void)ws_size;
    const float* x  = (const float*)d_in[0];
    const float* W1 = (const float*)d_in[1];
    const float* b1 = (const float*)d_in[2];
    const float* W2 = (const float*)d_in[3];
    const float* b2 = (const float*)d_in[4];
    const float* W3 = (const float*)d_in[5];
    // b3 (d_in[6]) vanishes under differentiation.

    float* ws = (float*)d_ws;
    size_t off = 0;
    auto carve = [&](size_t n) { float* p = ws + off; off += n; return p; };
    float* h1     = carve(HID);
    float* s1     = carve(HID);
    float* u      = carve(HID);
    float* cv     = carve(HID);
    float* d1v    = carve(HID);
    float* g1     = carve(HID);
    float* dxq    = carve(FREE);
    float* rowsum = carve(FREE);
    float* alpha  = carve(16);
    float* rhs    = carve(FREE);
    float* Cm     = carve((size_t)DIM * HID);    // C = (W1*s1)@W2, 2048x4096
    float* Hv     = carve((size_t)FREE * DIM);   // Hessian rows 1024..2047
    float* S      = carve((size_t)FREE * FREE);
    float* Xa     = carve((size_t)FREE * FREE);
    float* Xb     = carve((size_t)FREE * FREE);
    float* Tm     = carve((size_t)FREE * FREE);
    float* W1s    = carve((size_t)DIM * HID);    // pre-scaled A operands (reused)
    float* A1s    = W1s;                          // 1024x4096  (after C is built)
    float* A2s    = W1s + (size_t)FREE * HID;     // 1024x4096

    k_layer1<<<HID / 256, 256, 0, stream>>>(x, W1, b1, h1, s1);
    k_layer2<<<HID / 256, 256, 0, stream>>>(h1, W2, b2, W3, u, cv);
    k_g2<<<HID, 256, 0, stream>>>(W2, u, s1, d1v, g1);
    k_dx<<<FREE, 256, 0, stream>>>(W1, g1, dxq);

    // W1s = W1 ⊙_cols s1 ; C = W1s @ W2   [2048x4096x4096]
    k_colscale<<<(DIM * HID) / 256, 256, 0, stream>>>(W1, s1, W1s, HID, DIM * HID);
    k_gemm_rr_async<<<dim3(HID / TN, DIM / TM), 128, 0, stream>>>(W1s, HID, W2, HID, Cm, HID, HID);

    // A1s = W1[1024:] ⊙ d1 ; A2s = C[1024:] ⊙ cvec
    k_colscale<<<(FREE * HID) / 256, 256, 0, stream>>>(W1 + (size_t)FREE * HID, d1v, A1s, HID, FREE * HID);
    k_colscale<<<(FREE * HID) / 256, 256, 0, stream>>>(Cm + (size_t)FREE * HID, cv, A2s, HID, FREE * HID);

    // Hv = A1s @ W1^T ; Hv += A2s @ C^T   [1024x2048, K=4096]
    k_gemm_abt_async<<<dim3(DIM / TN, FREE / TM), 128, 0, stream>>>(
        A1s, HID, W1, HID, nullptr, 0, 0.f, 1.f, Hv, DIM, HID);
    k_gemm_abt_async<<<dim3(DIM / TN, FREE / TM), 128, 0, stream>>>(
        A2s, HID, Cm, HID, Hv, DIM, 1.f, 1.f, Hv, DIM, HID);

    k_rhs<<<FREE, 256, 0, stream>>>(Hv, x, dxq, rhs);
    k_extractS<<<FREE, 256, 0, stream>>>(Hv, S, rowsum);
    k_alpha<<<1, 256, 0, stream>>>(rowsum, alpha);
    k_scaleX0<<<(FREE * FREE) / 256, 256, 0, stream>>>(S, alpha, Xa);

    // Newton-Schulz (all operands symmetric -> pure A@B^T form):
    //   U = X @ S^T ; Xn = 2X - U @ X^T
    float* X  = Xa;
    float* Xn = Xb;
    for (int it = 0; it < NS_ITERS; ++it) {
        k_gemm_abt_async<<<dim3(FREE / TN, FREE / TM), 128, 0, stream>>>(
            X, FREE, S, FREE, nullptr, 0, 0.f, 1.f, Tm, FREE, FREE);
        k_gemm_abt_async<<<dim3(FREE / TN, FREE / TM), 128, 0, stream>>>(
            Tm, FREE, X, FREE, X, FREE, 2.f, -1.f, Xn, FREE, FREE);
        float* t = X; X = Xn; Xn = t;
    }

    k_out<<<FREE, 256, 0, stream>>>(X, rhs, (float*)d_out);
}